// DTCWTLayerBiort_30253749633561
// MI455X (gfx1250) — compile-verified
//
#include <hip/hip_runtime.h>

typedef float v2f __attribute__((ext_vector_type(2)));
typedef float v8f __attribute__((ext_vector_type(8)));

#define IMG_H 256
#define IMG_W 256
#define XP    24                 // K-padded patch width (22 -> 24, 6 chunks of K=4)
#define PATCH 22                 // 16 + 2*3 halo
#define WV_X   (XP * XP)         // 576 floats: input patch
#define WV_S   (XP * 16)         // 384 floats: one row-filtered plane (Lo or Hi)
#define WV_TOT (WV_X + 2 * WV_S) // 1344 floats per wave
#define WAVES  8
#define NPLANES 512              // 8 batch * 64 channels
#define TILES_PER_PLANE 256      // (256/16)^2

// half-sample symmetric extension: -1 -> 0, -2 -> 1, 256 -> 255, 257 -> 254
__device__ __forceinline__ int symmap(int g, int n) {
  if (g < 0)  g = -1 - g;
  if (g >= n) g = 2 * n - 1 - g;
  return g;
}

__global__ __launch_bounds__(256)
void dtcwt_level1_fused(const float* __restrict__ X,
                        const float* __restrict__ h0o,   // 5 taps, m=2
                        const float* __restrict__ h1o,   // 7 taps, m=3
                        float* __restrict__ out) {
  __shared__ float lds[WAVES * WV_TOT];
  __shared__ float sh0[8];
  __shared__ float sh1[8];

  const int tid  = threadIdx.x;
  const int wave = tid >> 5;
  const int lane = tid & 31;

  if (tid < 5) sh0[tid] = h0o[tid];
  if (tid < 7) sh1[tid] = h1o[tid];

  float* Xl  = &lds[wave * WV_TOT];   // [24][24] input patch (zero-padded K)
  float* SLo = Xl + WV_X;             // [24][16] row-filtered lowpass (rows rel 0..21 valid)
  float* SHi = SLo + WV_S;            // [24][16] row-filtered highpass

  const int gtile = blockIdx.x * WAVES + wave;            // 131072 tiles total
  const int plane = gtile >> 8;                           // which (n, c) image
  const int trem  = gtile & (TILES_PER_PLANE - 1);
  const int r0 = (trem >> 4) << 4;                        // tile origin in image
  const int c0 = (trem & 15) << 4;

  const float* Xp = X + (size_t)plane * (IMG_H * IMG_W);

  // ---- load 22x22 patch (global rows/cols r0-3..r0+18), zero-pad to 24x24 ----
  for (int i = lane; i < XP * XP; i += 32) {
    const int rr = i / XP, cc = i - rr * XP;
    float v = 0.f;
    if (rr < PATCH && cc < PATCH) {
      const int gr = symmap(r0 - 3 + rr, IMG_H);
      const int gc = symmap(c0 - 3 + cc, IMG_W);
      v = Xp[gr * IMG_W + gc];
    }
    Xl[i] = v;
  }
  // zero S pad rows 22,23 (read as K-pad by stage-2 B fragments)
  if (lane < 32) {
    SLo[22 * 16 + lane] = 0.f;
    SHi[22 * 16 + lane] = 0.f;
  }
  __syncthreads();

  const int half = lane >> 4;   // fragment lane-half
  const int nn   = lane & 15;   // M (A frag) / N (B frag) / column index

  // ---- banded filter matrices as WMMA fragments ----
  // band0(k,i) = h0[k-i-1] (5 taps, center 2), band1(k,i) = h1[k-i] (7 taps, center 3)
  // Identical function serves as stage-1 B (i = out col) and stage-2 A (i = out row).
  v2f fb0[6], fb1[6];
#pragma unroll
  for (int q = 0; q < 6; ++q) {
#pragma unroll
    for (int j = 0; j < 2; ++j) {
      const int k  = 4 * q + j + 2 * half;
      const int t0 = k - nn - 1;
      const int t1 = k - nn;
      const int t0c = t0 < 0 ? 0 : (t0 > 4 ? 4 : t0);
      const int t1c = t1 < 0 ? 0 : (t1 > 6 ? 6 : t1);
      const float v0 = sh0[t0c];
      const float v1 = sh1[t1c];
      fb0[q][j] = (t0 >= 0 && t0 < 5) ? v0 : 0.f;
      fb1[q][j] = (t1 >= 0 && t1 < 7) ? v1 : 0.f;
    }
  }

  // ---- stage 1: row filter.  S[rel r][n] = sum_k X[rel r][k] * band(k, n) ----
  // Need rel rows 0..21 -> two overlapping 16-row blocks starting at rel 0 and 6.
#pragma unroll
  for (int blk = 0; blk < 2; ++blk) {
    const int rbase = blk * 6;
    v8f accLo = {};
    v8f accHi = {};
#pragma unroll
    for (int q = 0; q < 6; ++q) {
      v2f a;
#pragma unroll
      for (int j = 0; j < 2; ++j) {
        const int k = 4 * q + j + 2 * half;
        a[j] = Xl[(rbase + nn) * XP + k];        // A: M = nn (row), K = k
      }
      accLo = __builtin_amdgcn_wmma_f32_16x16x4_f32(false, a, false, fb0[q],
                                                    (short)0, accLo, false, false);
      accHi = __builtin_amdgcn_wmma_f32_16x16x4_f32(false, a, false, fb1[q],
                                                    (short)0, accHi, false, false);
    }
#pragma unroll
    for (int v = 0; v < 8; ++v) {                // D row m = v + 8*half, col nn
      const int rr = rbase + v + 8 * half;
      SLo[rr * 16 + nn] = accLo[v];
      SHi[rr * 16 + nn] = accHi[v];
    }
  }
  __syncthreads();

  // ---- stage 2: column filter.  D[m][n] = sum_k band(k, m) * S[k][n] ----
  v8f dLL = {}, dLH = {}, dHL = {}, dHH = {};
#pragma unroll
  for (int q = 0; q < 6; ++q) {
    v2f bL, bH;
#pragma unroll
    for (int j = 0; j < 2; ++j) {
      const int k = 4 * q + j + 2 * half;        // B: K = k, N = nn
      bL[j] = SLo[k * 16 + nn];
      bH[j] = SHi[k * 16 + nn];
    }
    dLL = __builtin_amdgcn_wmma_f32_16x16x4_f32(false, fb0[q], false, bL,
                                                (short)0, dLL, false, false);
    dLH = __builtin_amdgcn_wmma_f32_16x16x4_f32(false, fb1[q], false, bL,
                                                (short)0, dLH, false, false);
    dHL = __builtin_amdgcn_wmma_f32_16x16x4_f32(false, fb0[q], false, bH,
                                                (short)0, dHL, false, false);
    dHH = __builtin_amdgcn_wmma_f32_16x16x4_f32(false, fb1[q], false, bH,
                                                (short)0, dHH, false, false);
  }

  // ---- outputs: LoLo full-res + six q2c subbands at half-res ----
  float* const oLL  = out;
  float* const oYhr = out + (size_t)NPLANES * IMG_H * IMG_W;        // +33,554,432
  float* const oYhi = oYhr + (size_t)NPLANES * 6 * 128 * 128;       // +50,331,648

  const size_t pLL = (size_t)plane * (IMG_H * IMG_W);
#pragma unroll
  for (int v = 0; v < 8; ++v) {
    const int m = v + 8 * half;
    oLL[pLL + (size_t)(r0 + m) * IMG_W + (c0 + nn)] = dLL[v];
  }

  const float is2 = 0.70710678118654752f;   // 1/sqrt(2)
  const int R0 = r0 >> 1, C0 = c0 >> 1;
  const int J  = nn >> 1;                   // quad column within tile
  const int p  = nn & 1;                    // 0: holds a,c   1: holds b,d
  const size_t sbs = (size_t)128 * 128;
  const size_t pSB = (size_t)plane * 6 * sbs;

  // q2c: a=y[2I][2J] b=y[2I][2J+1] c=y[2I+1][2J] d=y[2I+1][2J+1]
  // (sb1_r, sb1_i, sb2_r, sb2_i) = (a-d, b+c, a+d, b-c) * 1/sqrt2
  auto q2c_store = [&](const v8f d, int sb1, int sb2) {
#pragma unroll
    for (int iv = 0; iv < 4; ++iv) {
      const float x0 = d[2 * iv];       // row 2I (within lane-half)
      const float x1 = d[2 * iv + 1];   // row 2I+1
      const float y0 = __shfl_xor(x0, 1, 32);   // column partner
      const float y1 = __shfl_xor(x1, 1, 32);
      const float a  = p ? y0 : x0;
      const float b  = p ? x0 : y0;
      const float c  = p ? y1 : x1;
      const float dd = p ? x1 : y1;
      const int I = iv + 4 * half;
      const size_t off = (size_t)(R0 + I) * 128 + (C0 + J);
      if (p == 0) {   // even lanes store real parts
        oYhr[pSB + (size_t)sb1 * sbs + off] = (a - dd) * is2;
        oYhr[pSB + (size_t)sb2 * sbs + off] = (a + dd) * is2;
      } else {        // odd lanes store imaginary parts
        oYhi[pSB + (size_t)sb1 * sbs + off] = (b + c) * is2;
        oYhi[pSB + (size_t)sb2 * sbs + off] = (b - c) * is2;
      }
    }
  };
  q2c_store(dLH, 0, 5);   // LoHi -> d15, d165
  q2c_store(dHH, 1, 4);   // HiHi -> d45, d135
  q2c_store(dHL, 2, 3);   // HiLo -> d75, d105
}

extern "C" void kernel_launch(void* const* d_in, const int* in_sizes, int n_in,
                              void* d_out, int out_size, void* d_ws, size_t ws_size,
                              hipStream_t stream) {
  (void)in_sizes; (void)n_in; (void)out_size; (void)d_ws; (void)ws_size;
  const float* X  = (const float*)d_in[0];
  const float* h0 = (const float*)d_in[1];
  const float* h1 = (const float*)d_in[2];
  float* out = (float*)d_out;

  const int total_tiles = NPLANES * TILES_PER_PLANE;   // 131072
  const int blocks = total_tiles / WAVES;              // 16384
  hipLaunchKernelGGL(dtcwt_level1_fused, dim3(blocks), dim3(256), 0, stream,
                     X, h0, h1, out);
}